// HGLA_76510547411422
// MI455X (gfx1250) — compile-verified
//
#include <hip/hip_runtime.h>
#include <cstdint>

#define NBIN   256
#define NCHUNK 2048
#define NEG_SLOPE 0.2f

typedef __attribute__((ext_vector_type(2))) float v2f;
typedef __attribute__((ext_vector_type(8))) float v8f;

// Order-preserving float <-> uint encoding (for atomicMax on floats / radix keys)
__device__ __forceinline__ unsigned encf(float f) {
  unsigned u = __float_as_uint(f);
  return (u & 0x80000000u) ? ~u : (u | 0x80000000u);
}
__device__ __forceinline__ float decf(unsigned u) {
  unsigned b = (u & 0x80000000u) ? (u & 0x7FFFFFFFu) : ~u;
  return __uint_as_float(b);
}

// ---------------------------------------------------------------------------
// 0. workspace init
// ---------------------------------------------------------------------------
__global__ __launch_bounds__(256) void initWsKernel(float* deg, unsigned* gmax,
                                                    float* acc, float* den,
                                                    float* num, unsigned* gM, int N) {
  int i = blockIdx.x * blockDim.x + threadIdx.x;
  if (i < N) { deg[i] = 0.f; gmax[i] = 0u; acc[i] = 0.f; den[i] = 0.f; num[i] = 0.f; }
  if (i < 4) gM[i] = 0u;   // [0]=enc max gcn, [1]=enc max gat, [2]=sumexp gcn, [3]=sumexp gat
}

// ---------------------------------------------------------------------------
// 1. Fused triple mat-vec via f32 WMMA 16x16x4:
//    D[:,0]=x@W_gcn  D[:,1]=x@W_gat  D[:,2]=rowsum(x)
//    One wave handles a 16-row tile of x; C/4 WMMA steps of K=4.
//    B fragments (lane,step) are tile-invariant -> precomputed into LDS once
//    per block so the hot loop is: global b64 (A) + ds b64 (B) + v_wmma.
// ---------------------------------------------------------------------------
__global__ __launch_bounds__(256) void scoreMatvecKernel(
    const float* __restrict__ x, const float* __restrict__ Wg,
    const float* __restrict__ Wa, float* __restrict__ h, float* __restrict__ z,
    float* __restrict__ rs, int C, int ntiles) {
  __shared__ v2f sB[2048];           // (C/4)*32 entries, supports C <= 256
  int nsteps = C >> 2;
  for (int idx = threadIdx.x; idx < nsteps * 32; idx += blockDim.x) {
    int s  = idx >> 5;
    int ln = idx & 31;
    int mm = ln & 15;
    int hh = ln >> 4;
    int kr = s * 4 + 2 * hh;
    v2f b;
    b.x = (mm == 0) ? Wg[kr]     : (mm == 1) ? Wa[kr]     : (mm == 2) ? 1.0f : 0.0f;
    b.y = (mm == 0) ? Wg[kr + 1] : (mm == 1) ? Wa[kr + 1] : (mm == 2) ? 1.0f : 0.0f;
    sB[idx] = b;
  }
  __syncthreads();

  int wave = threadIdx.x >> 5;
  int lane = threadIdx.x & 31;
  int tile = blockIdx.x * 8 + wave;
  if (tile >= ntiles) return;

  int m  = lane & 15;   // row within tile (A) / column index n (B,C,D)
  int hi = lane >> 4;   // lane-half selects K pair
  const float* rowp = x + (size_t)(tile * 16 + m) * C + 2 * hi;
  const v2f* bp = sB + lane;

  v8f acc = {};
  for (int kk = 0; kk < C; kk += 4) {
    if ((kk & 63) == 0) __builtin_prefetch(rowp + kk + 64, 0, 0);
    // A fragment: v0 = K(0|2), v1 = K(1|3)  (ISA 32-bit 16x4 layout), 8B aligned
    v2f a = *(const v2f*)(rowp + kk);
    v2f b = bp[(kk >> 2) * 32];
    acc = __builtin_amdgcn_wmma_f32_16x16x4_f32(false, a, false, b, (short)0, acc,
                                                false, false);
  }

  // C/D layout: vgpr j, lanes 0-15 -> M=j (N=lane); lanes 16-31 -> M=8+j
  if (m < 3) {
    float* dst = (m == 0) ? h : (m == 1) ? z : rs;
    int base = tile * 16 + hi * 8;
#pragma unroll
    for (int j = 0; j < 8; j++) dst[base + j] = acc[j];
  }
}

// scalar tail for N not a multiple of 16 (not used when N=100000)
__global__ void scoreTailKernel(const float* __restrict__ x, const float* __restrict__ Wg,
                                const float* __restrict__ Wa, float* h, float* z,
                                float* rs, int N, int C, int start) {
  int row = start + blockIdx.x * blockDim.x + threadIdx.x;
  if (row >= N) return;
  float a = 0.f, b = 0.f, c = 0.f;
  const float* xp = x + (size_t)row * C;
  for (int i = 0; i < C; i++) { float v = xp[i]; a += v * Wg[i]; b += v * Wa[i]; c += v; }
  h[row] = a; z[row] = b; rs[row] = c;
}

// ---------------------------------------------------------------------------
// 2. Edge pass A: degree counts + GAT segment-max (self loops appended)
// ---------------------------------------------------------------------------
__global__ __launch_bounds__(256) void edgePassAKernel(
    const int* __restrict__ ei, int E, int N, const float* __restrict__ z,
    const float* __restrict__ asr, const float* __restrict__ adt, float* deg,
    unsigned* gmax) {
  int i = blockIdx.x * blockDim.x + threadIdx.x;
  int T = E + N;
  if (i >= T) return;
  int s_, d_;
  if (i < E) { s_ = ei[i]; d_ = ei[E + i]; } else { s_ = i - E; d_ = s_; }
  atomicAdd(&deg[d_], 1.0f);
  float e = asr[0] * z[s_] + adt[0] * z[d_];
  e = (e > 0.f) ? e : NEG_SLOPE * e;
  atomicMax(&gmax[d_], encf(e));
}

// ---------------------------------------------------------------------------
// 3. Edge pass B: GCN normalized aggregation + GAT exp-sum / weighted sum
// ---------------------------------------------------------------------------
__global__ __launch_bounds__(256) void edgePassBKernel(
    const int* __restrict__ ei, int E, int N, const float* __restrict__ h,
    const float* __restrict__ z, const float* __restrict__ asr,
    const float* __restrict__ adt, const float* __restrict__ deg,
    const unsigned* __restrict__ gmax, float* acc, float* den, float* num) {
  int i = blockIdx.x * blockDim.x + threadIdx.x;
  int T = E + N;
  if (i >= T) return;
  int s_, d_;
  if (i < E) { s_ = ei[i]; d_ = ei[E + i]; } else { s_ = i - E; d_ = s_; }
  float dis = rsqrtf(fmaxf(deg[s_], 1e-12f));
  float did = rsqrtf(fmaxf(deg[d_], 1e-12f));
  atomicAdd(&acc[d_], h[s_] * dis * did);
  float e = asr[0] * z[s_] + adt[0] * z[d_];
  e = (e > 0.f) ? e : NEG_SLOPE * e;
  float ex = __expf(e - decf(gmax[d_]));
  atomicAdd(&den[d_], ex);
  atomicAdd(&num[d_], z[s_] * ex);
}

// ---------------------------------------------------------------------------
// 4. Per-node finish + global max for the two softmaxes
// ---------------------------------------------------------------------------
__global__ __launch_bounds__(256) void nodeFinishKernel(
    const float* acc, const float* num, const float* den, const float* bg,
    const float* ba, float* g1o, float* g2o, unsigned* gM, int N) {
  int i = blockIdx.x * blockDim.x + threadIdx.x;
  int t = threadIdx.x;
  unsigned e1 = 0u, e2u = 0u;
  if (i < N) {
    float g1 = acc[i] + bg[0];
    float g2 = num[i] / den[i] + ba[0];
    g1o[i] = g1; g2o[i] = g2;
    e1 = encf(g1); e2u = encf(g2);
  }
  __shared__ unsigned s1[256];
  __shared__ unsigned s2[256];
  s1[t] = e1; s2[t] = e2u;
  __syncthreads();
  for (int d = 128; d > 0; d >>= 1) {
    if (t < d) {
      s1[t] = (s1[t] > s1[t + d]) ? s1[t] : s1[t + d];
      s2[t] = (s2[t] > s2[t + d]) ? s2[t] : s2[t + d];
    }
    __syncthreads();
  }
  if (t == 0) { atomicMax(&gM[0], s1[0]); atomicMax(&gM[1], s2[0]); }
}

__global__ __launch_bounds__(256) void sumExpKernel(const float* g1o, const float* g2o,
                                                    unsigned* gM, int N) {
  int i = blockIdx.x * blockDim.x + threadIdx.x;
  int t = threadIdx.x;
  float M1 = decf(gM[0]), M2 = decf(gM[1]);
  float e1 = 0.f, e2 = 0.f;
  if (i < N) { e1 = __expf(g1o[i] - M1); e2 = __expf(g2o[i] - M2); }
  __shared__ float s1[256];
  __shared__ float s2[256];
  s1[t] = e1; s2[t] = e2;
  __syncthreads();
  for (int d = 128; d > 0; d >>= 1) {
    if (t < d) { s1[t] += s1[t + d]; s2[t] += s2[t + d]; }
    __syncthreads();
  }
  if (t == 0) { atomicAdd((float*)&gM[2], s1[0]); atomicAdd((float*)&gM[3], s2[0]); }
}

// ---------------------------------------------------------------------------
// 5. scores + radix keys (descending => sort ascending on ~enc(score); stable)
// ---------------------------------------------------------------------------
__global__ __launch_bounds__(256) void scoresKeysKernel(
    const float* rs, const float* g1o, const float* g2o, const unsigned* gM,
    unsigned* keyA, int* idxA, int N) {
  int i = blockIdx.x * blockDim.x + threadIdx.x;
  if (i >= N) return;
  float M1 = decf(gM[0]), M2 = decf(gM[1]);
  float S1 = ((const float*)gM)[2];
  float S2 = ((const float*)gM)[3];
  float att = fmaxf(__expf(g1o[i] - M1) / S1, __expf(g2o[i] - M2) / S2);
  float sc = rs[i] + att;
  keyA[i] = ~encf(sc);
  idxA[i] = i;
}

// ---------------------------------------------------------------------------
// 6. 4-pass LSD radix sort of (key, idx), stable chunked counting sort
// ---------------------------------------------------------------------------
__global__ __launch_bounds__(256) void histKernel(const unsigned* __restrict__ ks,
                                                  unsigned* counts, int N, int chunk,
                                                  int shift) {
  int t = blockIdx.x * blockDim.x + threadIdx.x;
  if (t >= NCHUNK) return;
  for (int b = 0; b < NBIN; b++) counts[(size_t)b * NCHUNK + t] = 0u;
  int beg = t * chunk, end = beg + chunk;
  if (end > N) end = N;
  for (int i = beg; i < end; i++) {
    unsigned bin = (ks[i] >> shift) & 255u;
    counts[(size_t)bin * NCHUNK + t]++;
  }
}

__global__ __launch_bounds__(256) void binScanKernel(unsigned* counts, unsigned* binTotal) {
  int b = blockIdx.x, t = threadIdx.x;
  unsigned* col = counts + (size_t)b * NCHUNK;
  unsigned loc[8];
  unsigned s = 0;
#pragma unroll
  for (int j = 0; j < 8; j++) { loc[j] = col[t * 8 + j]; s += loc[j]; }
  __shared__ unsigned sh[256];
  sh[t] = s;
  __syncthreads();
  for (int d = 1; d < 256; d <<= 1) {
    unsigned v = (t >= d) ? sh[t - d] : 0u;
    __syncthreads();
    sh[t] += v;
    __syncthreads();
  }
  unsigned run = (t == 0) ? 0u : sh[t - 1];
  if (t == 255) binTotal[b] = sh[255];
#pragma unroll
  for (int j = 0; j < 8; j++) { unsigned c = loc[j]; col[t * 8 + j] = run; run += c; }
}

__global__ __launch_bounds__(256) void binOffsetScanKernel(const unsigned* binTotal,
                                                           unsigned* binOffset) {
  int t = threadIdx.x;
  __shared__ unsigned sh[256];
  sh[t] = binTotal[t];
  __syncthreads();
  for (int d = 1; d < 256; d <<= 1) {
    unsigned v = (t >= d) ? sh[t - d] : 0u;
    __syncthreads();
    sh[t] += v;
    __syncthreads();
  }
  binOffset[t] = (t == 0) ? 0u : sh[t - 1];
}

__global__ __launch_bounds__(256) void scatterKernel(const unsigned* __restrict__ ks,
                                                     const int* __restrict__ is,
                                                     unsigned* kd, int* id,
                                                     unsigned* counts,
                                                     const unsigned* __restrict__ binOffset,
                                                     int N, int chunk, int shift) {
  int t = blockIdx.x * blockDim.x + threadIdx.x;
  if (t >= NCHUNK) return;
  int beg = t * chunk, end = beg + chunk;
  if (end > N) end = N;
  for (int i = beg; i < end; i++) {
    unsigned key = ks[i];
    unsigned bin = (key >> shift) & 255u;
    unsigned off = counts[(size_t)bin * NCHUNK + t];
    counts[(size_t)bin * NCHUNK + t] = off + 1u;
    unsigned pos = binOffset[bin] + off;
    kd[pos] = key;
    id[pos] = is[i];
  }
}

// ---------------------------------------------------------------------------
// 7. outputs: mapping, batch gather, x gather (float4), edge remap + valid
// ---------------------------------------------------------------------------
__global__ __launch_bounds__(256) void mappingInitKernel(int* mapping, int N) {
  int i = blockIdx.x * blockDim.x + threadIdx.x;
  if (i < N) mapping[i] = -1;
}

__global__ __launch_bounds__(256) void mappingSetKernel(const int* __restrict__ topk,
                                                        const int* __restrict__ batch,
                                                        int* mapping, float* bo, int k) {
  int j = blockIdx.x * blockDim.x + threadIdx.x;
  if (j >= k) return;
  int t = topk[j];
  mapping[t] = j;
  bo[j] = (float)batch[t];
}

__global__ __launch_bounds__(256) void gatherXKernel(const float* __restrict__ x,
                                                     const int* __restrict__ topk,
                                                     float* __restrict__ xo, int k, int C4) {
  long tid = (long)blockIdx.x * blockDim.x + threadIdx.x;
  long tot = (long)k * C4;
  if (tid >= tot) return;
  int j = (int)(tid / C4);
  int c = (int)(tid % C4);
  int row = topk[j];
  const float4* src = (const float4*)x;
  ((float4*)xo)[(size_t)j * C4 + c] = src[(size_t)row * C4 + c];
}

__global__ __launch_bounds__(256) void edgeOutKernel(const int* __restrict__ ei,
                                                     const int* __restrict__ mapping,
                                                     float* e2o, float* vo, int E) {
  int i = blockIdx.x * blockDim.x + threadIdx.x;
  if (i >= E) return;
  int a = mapping[ei[i]];
  int b = mapping[ei[E + i]];
  bool v = (a >= 0) && (b >= 0);
  e2o[i]     = v ? (float)a : -1.0f;
  e2o[E + i] = v ? (float)b : -1.0f;
  vo[i] = v ? 1.0f : 0.0f;
}

// ---------------------------------------------------------------------------
extern "C" void kernel_launch(void* const* d_in, const int* in_sizes, int n_in,
                              void* d_out, int out_size, void* d_ws, size_t ws_size,
                              hipStream_t stream) {
  const float* x   = (const float*)d_in[0];
  const int* ei    = (const int*)d_in[1];
  const int* batch = (const int*)d_in[2];
  const float* Wg  = (const float*)d_in[3];
  const float* bg  = (const float*)d_in[4];
  const float* Wa  = (const float*)d_in[5];
  const float* ba  = (const float*)d_in[6];
  const float* asr = (const float*)d_in[7];
  const float* adt = (const float*)d_in[8];

  int N = in_sizes[2];
  int E = in_sizes[1] / 2;
  int C = in_sizes[3];
  int k = (N + 1) / 2;

  float* F = (float*)d_ws;
  float* h    = F + 0 * (size_t)N;
  float* z    = F + 1 * (size_t)N;
  float* rs   = F + 2 * (size_t)N;
  float* deg  = F + 3 * (size_t)N;
  float* acc  = F + 4 * (size_t)N;
  float* den  = F + 5 * (size_t)N;
  float* num  = F + 6 * (size_t)N;
  float* g1o  = F + 7 * (size_t)N;
  float* g2o  = F + 8 * (size_t)N;
  unsigned* gmax = (unsigned*)(F + 9 * (size_t)N);
  unsigned* keyA = (unsigned*)(F + 10 * (size_t)N);
  int*      idxA = (int*)(F + 11 * (size_t)N);
  unsigned* keyB = (unsigned*)(F + 12 * (size_t)N);
  int*      idxB = (int*)(F + 13 * (size_t)N);
  int*   mapping = (int*)(F + 14 * (size_t)N);
  unsigned* counts   = (unsigned*)(F + 15 * (size_t)N);
  unsigned* binTotal = counts + (size_t)NBIN * NCHUNK;
  unsigned* binOffset = binTotal + NBIN;
  unsigned* gM = binOffset + NBIN;   // 4 words

  float* xo  = (float*)d_out;
  float* e2o = xo + (size_t)k * C;
  float* bo  = e2o + 2 * (size_t)E;
  float* vo  = bo + k;

  int nb = (N + 255) / 256;
  initWsKernel<<<nb, 256, 0, stream>>>(deg, gmax, acc, den, num, gM, N);

  int ntiles = N / 16;
  if (ntiles > 0) {
    int blocks = (ntiles + 7) / 8;
    scoreMatvecKernel<<<blocks, 256, 0, stream>>>(x, Wg, Wa, h, z, rs, C, ntiles);
  }
  if (ntiles * 16 < N) {
    scoreTailKernel<<<1, 256, 0, stream>>>(x, Wg, Wa, h, z, rs, N, C, ntiles * 16);
  }

  int T = E + N;
  int eb = (T + 255) / 256;
  edgePassAKernel<<<eb, 256, 0, stream>>>(ei, E, N, z, asr, adt, deg, gmax);
  edgePassBKernel<<<eb, 256, 0, stream>>>(ei, E, N, h, z, asr, adt, deg, gmax, acc, den, num);

  nodeFinishKernel<<<nb, 256, 0, stream>>>(acc, num, den, bg, ba, g1o, g2o, gM, N);
  sumExpKernel<<<nb, 256, 0, stream>>>(g1o, g2o, gM, N);
  scoresKeysKernel<<<nb, 256, 0, stream>>>(rs, g1o, g2o, gM, keyA, idxA, N);

  int chunk = (N + NCHUNK - 1) / NCHUNK;
  unsigned* ks = keyA; int* is = idxA;
  unsigned* kd = keyB; int* id = idxB;
  for (int p = 0; p < 4; p++) {
    histKernel<<<NCHUNK / 256, 256, 0, stream>>>(ks, counts, N, chunk, 8 * p);
    binScanKernel<<<NBIN, 256, 0, stream>>>(counts, binTotal);
    binOffsetScanKernel<<<1, 256, 0, stream>>>(binTotal, binOffset);
    scatterKernel<<<NCHUNK / 256, 256, 0, stream>>>(ks, is, kd, id, counts, binOffset, N,
                                                    chunk, 8 * p);
    unsigned* tk = ks; ks = kd; kd = tk;
    int* ti = is; is = id; id = ti;
  }
  // after 4 passes, sorted (ascending on ~enc(score) == descending score) back in keyA/idxA

  mappingInitKernel<<<nb, 256, 0, stream>>>(mapping, N);
  int kb = (k + 255) / 256;
  mappingSetKernel<<<kb, 256, 0, stream>>>(is, batch, mapping, bo, k);

  int C4 = C / 4;
  long tot = (long)k * C4;
  int gb = (int)((tot + 255) / 256);
  gatherXKernel<<<gb, 256, 0, stream>>>(x, is, xo, k, C4);

  int ebE = (E + 255) / 256;
  edgeOutKernel<<<ebE, 256, 0, stream>>>(ei, mapping, e2o, vo, E);
}